// CategoryAlign_Module_8675833938581
// MI455X (gfx1250) — compile-verified
//
#include <hip/hip_runtime.h>
#include <hip/hip_bf16.h>

// Problem constants (match reference setup_inputs)
#define KCLS   19
#define MPAD   32
#define CCH    256
#define HWTOT  16384
#define SPLITK 8
#define CHUNK  (HWTOT / SPLITK)   // 2048 hw elements per split
#define TK     64                 // hw elements staged in LDS per stage
#define LSTR   68                 // padded LDS row stride (floats): 68 % 64 banks = 4
#define NSTAGE (CHUNK / TK)       // 32 stages

typedef float v2f __attribute__((ext_vector_type(2)));
typedef float v8f __attribute__((ext_vector_type(8)));
typedef int   v4i __attribute__((ext_vector_type(4)));

// ---- CDNA5 async Global->LDS path (guarded; falls back to ld+ds_store) ----
#if defined(__has_builtin)
# if __has_builtin(__builtin_amdgcn_global_load_async_to_lds_b128)
#  define HAVE_ASYNC_LDS 1
# endif
#endif
#ifndef HAVE_ASYNC_LDS
# define HAVE_ASYNC_LDS 0
#endif

// Pointer-to-AS1(v4i) and pointer-to-AS3(v4i), per the builtin's signature:
// 'int __attribute__((vector_size(16))) __device__ *' etc.
typedef __attribute__((address_space(1))) v4i* as1_v4i_p;
typedef __attribute__((address_space(3))) v4i* as3_v4i_p;

// Copy 16 bytes global -> LDS (per lane).
__device__ __forceinline__ void copy16_g2l(const float* g, float* l) {
#if HAVE_ASYNC_LDS
    __builtin_amdgcn_global_load_async_to_lds_b128(
        (as1_v4i_p)(void*)g, (as3_v4i_p)(void*)l, /*imm offset=*/0, /*cpol=*/0);
#else
    *(float4*)l = *(const float4*)g;
#endif
}

// Wait for staged data to land in LDS, then workgroup barrier.
__device__ __forceinline__ void wait_staging() {
#if HAVE_ASYNC_LDS
# if defined(__has_builtin) && __has_builtin(__builtin_amdgcn_s_wait_asynccnt)
    __builtin_amdgcn_s_wait_asynccnt(0);
# else
    asm volatile("s_wait_asynccnt 0" ::: "memory");
# endif
#endif
    __syncthreads();
}

// ---------------------------------------------------------------------------
// Kernel A: split-K batched GEMM  vec[b,k,c] = sum_p preds[b,k,p]*feats[b,c,p]
// plus folded row sums psum[b,k] = sum_p preds[b,k,p].
// V_WMMA_F32_16X16X4_F32 (full f32 fidelity; workload is HBM-bound).
// Double-buffered LDS: async-stage tile it+1 while WMMA-consuming tile it;
// one s_wait_asynccnt + barrier per stage.
// grid = (C/64, B, 2*SPLITK), block = 256 (8 waves, wave32).
// ---------------------------------------------------------------------------
__global__ __launch_bounds__(256) void gemm_vec_kernel(
    const float* __restrict__ preds1, const float* __restrict__ feats1,
    const float* __restrict__ preds2, const float* __restrict__ feats2,
    float* __restrict__ vecPart, float* __restrict__ psumPart)
{
    __shared__ float PtAll[2 * MPAD * LSTR];   // 2 x (32 x 64) class tiles
    __shared__ float FtAll[2 * 64 * LSTR];     // 2 x (64 x 64) channel tiles

    const int cblk  = blockIdx.x;            // 0..3  -> 64 channels
    const int b     = blockIdx.y;            // 0..7
    const int s     = blockIdx.z >> 3;       // stream 0/1
    const int split = blockIdx.z & (SPLITK - 1);

    const float* __restrict__ preds = s ? preds2 : preds1;
    const float* __restrict__ feats = s ? feats2 : feats1;

    const int tid  = threadIdx.x;
    const int lane = tid & 31;
    const int wv   = tid >> 5;               // 0..7
    const int mt   = wv >> 2;                // m tile 0/1 (classes 0-15 / 16-31)
    const int ns   = wv & 3;                 // n subtile 0..3 (16 channels each)

    const int chunk0 = split * CHUNK;

    // Cooperative-load indices
    const int pr = tid >> 3;                 // 0..31  (class row)
    const int pc = (tid & 7) * 8;            // 8 floats per thread
    const int fr = tid >> 2;                 // 0..63  (channel row)
    const int fc = (tid & 3) * 16;           // 16 floats per thread

    const size_t predRow = ((size_t)b * KCLS + (pr < KCLS ? pr : 0)) * HWTOT + chunk0 + pc;
    const size_t featRow = ((size_t)b * CCH + (size_t)cblk * 64 + fr) * HWTOT + chunk0 + fc;

    // Stage one 64-wide hw slice into LDS buffer bi.
    auto stage_load = [&](int st, int bi) {
        const int off = st * TK;
        float* Pb = &PtAll[bi * (MPAD * LSTR)];
        float* Fb = &FtAll[bi * (64 * LSTR)];
        if (pr < KCLS) {
            const float* g = preds + predRow + off;
            copy16_g2l(g + 0, &Pb[pr * LSTR + pc + 0]);
            copy16_g2l(g + 4, &Pb[pr * LSTR + pc + 4]);
        }
        const float* gf = feats + featRow + off;
        #pragma unroll
        for (int j = 0; j < 4; ++j)
            copy16_g2l(gf + 4 * j, &Fb[fr * LSTR + fc + 4 * j]);
    };

    v8f acc = {0.f, 0.f, 0.f, 0.f, 0.f, 0.f, 0.f, 0.f};
    float asum = 0.f;                        // folded preds row-sum partial

    // WMMA f32 16x16x4 fragment addressing (per ISA layout):
    // A: lanes 0-15 -> M=lane, K={0,1}; lanes 16-31 -> M=lane-16, K={2,3}
    // B: lanes 0-15 -> K={0,1}, N=lane; lanes 16-31 -> K={2,3}, N=lane-16
    const int m  = lane & 15;
    const int kb = (lane >> 4) * 2;
    const int aOff = (mt * 16 + m) * LSTR + kb;
    const int bOff = (ns * 16 + m) * LSTR + kb;

    // Prologue: stage 0 into buffer 0; zero-pad classes 19..31 once (both bufs).
    stage_load(0, 0);
    if (pr >= KCLS) {
        const float4 z = {0.f, 0.f, 0.f, 0.f};
        #pragma unroll
        for (int bi = 0; bi < 2; ++bi) {
            *(float4*)&PtAll[bi * (MPAD * LSTR) + pr * LSTR + pc + 0] = z;
            *(float4*)&PtAll[bi * (MPAD * LSTR) + pr * LSTR + pc + 4] = z;
        }
    }
    wait_staging();

    for (int it = 0; it < NSTAGE; ++it) {
        const int cur = it & 1;
        if (it + 1 < NSTAGE) stage_load(it + 1, cur ^ 1);  // async prefetch next

        const float* pA = &PtAll[cur * (MPAD * LSTR) + aOff];
        const float* pB = &FtAll[cur * (64 * LSTR) + bOff];
        #pragma unroll
        for (int kk = 0; kk < TK; kk += 4) {
            v2f a  = *(const v2f*)(pA + kk);
            v2f bm = *(const v2f*)(pB + kk);
            acc = __builtin_amdgcn_wmma_f32_16x16x4_f32(
                /*neg_a=*/false, a, /*neg_b=*/false, bm,
                /*c_mod=*/(short)0, acc, /*reuse_a=*/false, /*reuse_b=*/false);
            asum += a.x + a.y;               // preds mass (co-executes with WMMA)
        }
        wait_staging();
    }

    // psum partial: lanes 0-15 hold class m with K-offsets {0,1}; add upper half
    asum += __shfl_down(asum, 16);
    if (ns == 0 && lane < 16) {
        psumPart[(((size_t)s * SPLITK + split) * 8 + b) * MPAD + mt * 16 + lane] = asum;
    }

    // Store 16x16 f32 accumulator: VGPR v -> row v (lanes 0-15) / v+8 (16-31)
    const size_t obase = (((size_t)s * SPLITK + split) * 8 + b) * (size_t)MPAD * CCH;
    const int colg = cblk * 64 + ns * 16 + m;
    const int row0 = mt * 16 + (lane >> 4) * 8;
    #pragma unroll
    for (int v = 0; v < 8; ++v)
        vecPart[obase + (size_t)(row0 + v) * CCH + colg] = acc[v];
}

// ---------------------------------------------------------------------------
// Kernel B: reduce split-K partials, divide by class mass, L2-normalize over
// the class dim per (b,c) with eps, mean over batch -> context[2][19][256].
// grid = (2 streams), block = 256 (thread = channel c).
// ---------------------------------------------------------------------------
__global__ __launch_bounds__(256) void context_kernel(
    const float* __restrict__ vecPart, const float* __restrict__ psumPart,
    float* __restrict__ context)
{
    const int s = blockIdx.x;
    const int c = threadIdx.x;

    float ctx[KCLS];
    #pragma unroll
    for (int k = 0; k < KCLS; ++k) ctx[k] = 0.f;

    for (int b = 0; b < 8; ++b) {
        float vec[KCLS];
        float nrm2 = 0.f;
        for (int k = 0; k < KCLS; ++k) {
            float accv = 0.f, ps = 0.f;
            #pragma unroll
            for (int sp = 0; sp < SPLITK; ++sp) {
                const size_t sb = (((size_t)s * SPLITK + sp) * 8 + b);
                accv += vecPart[sb * (size_t)MPAD * CCH + (size_t)k * CCH + c];
                ps   += psumPart[sb * MPAD + k];
            }
            const float v = accv / ps;
            vec[k] = v;
            nrm2 += v * v;
        }
        const float inv = 1.0f / fmaxf(sqrtf(nrm2), 1e-12f);
        #pragma unroll
        for (int k = 0; k < KCLS; ++k) ctx[k] += vec[k] * inv * 0.125f;
    }

    #pragma unroll
    for (int k = 0; k < KCLS; ++k)
        context[((size_t)s * KCLS + k) * CCH + c] = ctx[k];
}

// ---------------------------------------------------------------------------
// Kernel C: Pearson matrix. One block; wave-per-row mean-center + normalize
// of the 38 context rows in LDS (wave32 shuffles), then 19x19 dot products.
// ---------------------------------------------------------------------------
__global__ __launch_bounds__(512) void pearson_kernel(
    const float* __restrict__ context, float* __restrict__ out)
{
    __shared__ float X[2 * KCLS * CCH];

    for (int i = threadIdx.x; i < 2 * KCLS * CCH; i += 512)
        X[i] = context[i];
    __syncthreads();

    const int lane = threadIdx.x & 31;
    const int wid  = threadIdx.x >> 5;     // 16 waves

    for (int r = wid; r < 2 * KCLS; r += 16) {
        float* row = &X[r * CCH];
        float s0 = 0.f;
        for (int i = lane; i < CCH; i += 32) s0 += row[i];
        #pragma unroll
        for (int off = 16; off > 0; off >>= 1) s0 += __shfl_down(s0, off);
        const float mean = __shfl(s0, 0) * (1.0f / (float)CCH);

        float s1 = 0.f;
        for (int i = lane; i < CCH; i += 32) {
            const float d = row[i] - mean;
            row[i] = d;
            s1 += d * d;
        }
        #pragma unroll
        for (int off = 16; off > 0; off >>= 1) s1 += __shfl_down(s1, off);
        const float inv = 1.0f / sqrtf(__shfl(s1, 0));
        for (int i = lane; i < CCH; i += 32) row[i] *= inv;
    }
    __syncthreads();

    for (int o = threadIdx.x; o < KCLS * KCLS; o += 512) {
        const int i = o / KCLS;
        const int j = o % KCLS;
        const float* xr = &X[i * CCH];
        const float* yr = &X[(KCLS + j) * CCH];
        float d = 0.f;
        for (int t = 0; t < CCH; ++t) d += xr[t] * yr[t];
        out[o] = d;
    }
}

// ---------------------------------------------------------------------------
extern "C" void kernel_launch(void* const* d_in, const int* in_sizes, int n_in,
                              void* d_out, int out_size, void* d_ws, size_t ws_size,
                              hipStream_t stream) {
    (void)in_sizes; (void)n_in; (void)out_size; (void)ws_size;

    const float* preds1 = (const float*)d_in[0];
    const float* feats1 = (const float*)d_in[1];
    const float* preds2 = (const float*)d_in[2];
    const float* feats2 = (const float*)d_in[3];
    float* out = (float*)d_out;

    // Workspace layout (floats): all regions fully rewritten every call.
    float* vecPart  = (float*)d_ws;                               // 2*8*8*32*256 = 1048576
    float* psumPart = vecPart + (size_t)2 * SPLITK * 8 * MPAD * CCH;
    float* context  = psumPart + (size_t)2 * SPLITK * 8 * MPAD;   // 2*19*256 = 9728

    dim3 gA(CCH / 64, 8, 2 * SPLITK);
    gemm_vec_kernel<<<gA, 256, 0, stream>>>(preds1, feats1, preds2, feats2,
                                            vecPart, psumPart);
    context_kernel<<<2, 256, 0, stream>>>(vecPart, psumPart, context);
    pearson_kernel<<<1, 512, 0, stream>>>(context, out);
}